// MixtureOfLogistics_10419590660479
// MI455X (gfx1250) — compile-verified
//
#include <hip/hip_runtime.h>

typedef __attribute__((ext_vector_type(2))) float v2f;
typedef __attribute__((ext_vector_type(8))) float v8f;

// Wave32 all-lane sum via V_WMMA_F32_16X16X4_F32 with B = ones.
// A 16x4 layout: lanes 0-15 are M=0..15; VGPR0 = K0 (lanes 0-15) / K2 (lanes 16-31),
// VGPR1 = K1 / K3.  With A.vgpr0 = x, A.vgpr1 = 0, B = ones:
//   D[m][n] = x_m + x_{m+16}   (all n)
// C/D layout: c[v] at lane L<16 is D[v][L], at lane L>=16 is D[v+8][L-16].
// Sum c[0..7] per lane -> lanes 0-15 hold sum(m=0..7), lanes 16-31 hold sum(m=8..15).
// One xor-16 shuffle adds the two halves: every lane gets the full 32-lane total.
__device__ __forceinline__ float wave_sum_wmma(float x) {
    v2f a; a[0] = x;    a[1] = 0.0f;
    v2f b; b[0] = 1.0f; b[1] = 1.0f;
    v8f c = {};
    c = __builtin_amdgcn_wmma_f32_16x16x4_f32(
            /*neg_a=*/false, a, /*neg_b=*/false, b,
            /*c_mod=*/(short)0, c, /*reuse_a=*/false, /*reuse_b=*/false);
    float p = ((c[0] + c[1]) + (c[2] + c[3])) + ((c[4] + c[5]) + (c[6] + c[7]));
    p += __shfl_xor(p, 16, 32);
    return p;
}

#define NEG_INF (-3.402823466e38f)

__global__ void __launch_bounds__(256)
mol_partial_kernel(const float* __restrict__ w,
                   const float* __restrict__ loc,
                   const float* __restrict__ sc,
                   const float* __restrict__ tgt,
                   float* __restrict__ partial,
                   long long nrows) {
    const long long stride = (long long)gridDim.x * (long long)blockDim.x;
    float acc = 0.0f;

    for (long long r = (long long)blockIdx.x * blockDim.x + threadIdx.x;
         r < nrows; r += stride) {
        const float4* wp = (const float4*)(w   + (size_t)r * 16);
        const float4* lp = (const float4*)(loc + (size_t)r * 16);
        const float4* sp = (const float4*)(sc  + (size_t)r * 16);
        const float x = tgt[r];

        float wv[16];
        float ev[16];
        float mw = NEG_INF, me = NEG_INF;

#pragma unroll
        for (int cch = 0; cch < 4; ++cch) {
            const float4 wq = wp[cch];
            const float4 lq = lp[cch];
            const float4 sq = sp[cch];
            const float w4[4] = {wq.x, wq.y, wq.z, wq.w};
            const float l4[4] = {lq.x, lq.y, lq.z, lq.w};
            const float s4[4] = {sq.x, sq.y, sq.z, sq.w};
#pragma unroll
            for (int j = 0; j < 4; ++j) {
                const int k = cch * 4 + j;
                const float wk = w4[j];
                const float s  = s4[j];
                const float rs = __builtin_amdgcn_rcpf(s);   // v_rcp_f32
                const float z  = (x - l4[j]) * rs;
                const float az = fabsf(z);
                const float t  = __expf(-az);                // v_exp_f32
                const float u  = 1.0f + t;
                // logistic logpdf: -|z| - 2*log1p(e^{-|z|}) - log(s)
                //                = -|z| - log((1+t)^2 * s)   (fused: one v_log_f32)
                const float e  = wk - az - __logf(u * u * s);
                wv[k] = wk;
                ev[k] = e;
                mw = fmaxf(mw, wk);
                me = fmaxf(me, e);
            }
        }

        float sw = 0.0f, se = 0.0f;
#pragma unroll
        for (int k = 0; k < 16; ++k) {
            sw += __expf(wv[k] - mw);
            se += __expf(ev[k] - me);
        }
        // log_prob = logsumexp(w + comp) - logsumexp(w)
        acc += (me + __logf(se)) - (mw + __logf(sw));
    }

    // Block reduction: 8 waves, each reduced through the WMMA matrix pipe.
    const float wsum = wave_sum_wmma(acc);
    __shared__ float lds[8];
    const int lane = threadIdx.x & 31;
    const int wave = threadIdx.x >> 5;
    if (lane == 0) lds[wave] = wsum;
    __syncthreads();
    if (threadIdx.x == 0) {
        float bsum = 0.0f;
#pragma unroll
        for (int i = 0; i < 8; ++i) bsum += lds[i];
        partial[blockIdx.x] = bsum;
    }
}

__global__ void __launch_bounds__(256)
mol_final_kernel(const float* __restrict__ partial,
                 float* __restrict__ out,
                 int np, float inv_n) {
    float acc = 0.0f;
    for (int i = threadIdx.x; i < np; i += 256) acc += partial[i];

    const float wsum = wave_sum_wmma(acc);
    __shared__ float lds[8];
    const int lane = threadIdx.x & 31;
    const int wave = threadIdx.x >> 5;
    if (lane == 0) lds[wave] = wsum;
    __syncthreads();
    if (threadIdx.x == 0) {
        float total = 0.0f;
#pragma unroll
        for (int i = 0; i < 8; ++i) total += lds[i];
        out[0] = total * inv_n;
    }
}

extern "C" void kernel_launch(void* const* d_in, const int* in_sizes, int n_in,
                              void* d_out, int out_size, void* d_ws, size_t ws_size,
                              hipStream_t stream) {
    const float* w   = (const float*)d_in[0];  // weight [B,T,K]
    const float* loc = (const float*)d_in[1];  // loc    [B,T,K]
    const float* sc  = (const float*)d_in[2];  // scale  [B,T,K]
    const float* tgt = (const float*)d_in[3];  // targets[B,T]
    float* out = (float*)d_out;
    float* partial = (float*)d_ws;             // 2048 floats = 8 KB scratch

    const long long nrows = (long long)in_sizes[3];  // B*T rows
    const int blocks = 2048, threads = 256;

    mol_partial_kernel<<<blocks, threads, 0, stream>>>(w, loc, sc, tgt, partial, nrows);
    mol_final_kernel<<<1, threads, 0, stream>>>(partial, out, blocks,
                                                1.0f / (float)nrows);
}